// xSpatioTemporalNN_40742059770456
// MI455X (gfx1250) — compile-verified
//
#include <hip/hip_runtime.h>
#include <math.h>

// Problem constants (from the reference)
#define NXD   2000   // NX
#define NTD   500    // NT
#define NPD   4      // NP
#define NZD   32     // NZ
#define BATCH 4096   // B
#define KDIM  80     // NZ*NR + NP*4 = 64 + 16

typedef __attribute__((ext_vector_type(2))) float v2f;
typedef __attribute__((ext_vector_type(8))) float v8f;

// ---------------------------------------------------------------------------
// Phase 1: per batch element, compute z_cat[80].
// One 256-thread block (8 wave32) per batch element. Each thread strides the
// n=0..1999 reduction, keeping 40 fp32 accumulators; wave32 shuffle butterfly
// reduce, then an 8-way LDS reduce. L2-bandwidth bound (all inputs ~160MB fit
// in the 192MB L2), so we use b128 loads + global_prefetch for the factors
// stream.
// ---------------------------------------------------------------------------
__global__ __launch_bounds__(256)
void zcat_kernel(const float* __restrict__ relations,   // [NX, 1, NX]
                 const float* __restrict__ exogenous,   // [NT, NX, NP]
                 const float* __restrict__ factors,     // [NT, NX, NZ]
                 const int*   __restrict__ t_idx,       // [B]
                 const int*   __restrict__ x_idx,       // [B]
                 float*       __restrict__ zcat)        // [B, 80]
{
    __shared__ float red[8 * 48];   // 8 waves x 40 partials (padded stride 48)

    const int b    = blockIdx.x;
    const int tid  = threadIdx.x;
    const int lane = tid & 31;
    const int wave = tid >> 5;

    const int t = t_idx[b];
    const int x = x_idx[b];

    const float* relrow = relations + (size_t)x * NXD;
    const float* fbase  = factors   + ((size_t)t) * NXD * NZD;
    const float* e0base = exogenous + ((size_t)t) * NXD * NPD;
    const float* e1base = exogenous + ((size_t)t + 1) * NXD * NPD;

    float accz[NZD];
    float acce0[NPD], acce1[NPD];
#pragma unroll
    for (int k = 0; k < NZD; ++k) accz[k] = 0.0f;
#pragma unroll
    for (int p = 0; p < NPD; ++p) { acce0[p] = 0.0f; acce1[p] = 0.0f; }

    for (int n = tid; n < NXD; n += 256) {
        // prefetch next factors row for this thread (gfx1250 global_prefetch_b8)
        if (n + 256 < NXD)
            __builtin_prefetch(fbase + (size_t)(n + 256) * NZD, 0, 1);

        const float r = relrow[n];

        const float4* f4 = (const float4*)(fbase + (size_t)n * NZD);
#pragma unroll
        for (int j = 0; j < NZD / 4; ++j) {
            float4 v = f4[j];
            accz[4 * j + 0] = fmaf(r, v.x, accz[4 * j + 0]);
            accz[4 * j + 1] = fmaf(r, v.y, accz[4 * j + 1]);
            accz[4 * j + 2] = fmaf(r, v.z, accz[4 * j + 2]);
            accz[4 * j + 3] = fmaf(r, v.w, accz[4 * j + 3]);
        }
        float4 e0 = *(const float4*)(e0base + (size_t)n * NPD);
        float4 e1 = *(const float4*)(e1base + (size_t)n * NPD);
        acce0[0] = fmaf(r, e0.x, acce0[0]);
        acce0[1] = fmaf(r, e0.y, acce0[1]);
        acce0[2] = fmaf(r, e0.z, acce0[2]);
        acce0[3] = fmaf(r, e0.w, acce0[3]);
        acce1[0] = fmaf(r, e1.x, acce1[0]);
        acce1[1] = fmaf(r, e1.y, acce1[1]);
        acce1[2] = fmaf(r, e1.z, acce1[2]);
        acce1[3] = fmaf(r, e1.w, acce1[3]);
    }

    // wave32 butterfly reduce each of the 40 accumulators
#pragma unroll
    for (int k = 0; k < NZD; ++k) {
        float v = accz[k];
#pragma unroll
        for (int off = 16; off > 0; off >>= 1) v += __shfl_xor(v, off, 32);
        accz[k] = v;
    }
#pragma unroll
    for (int p = 0; p < NPD; ++p) {
        float v0 = acce0[p], v1 = acce1[p];
#pragma unroll
        for (int off = 16; off > 0; off >>= 1) {
            v0 += __shfl_xor(v0, off, 32);
            v1 += __shfl_xor(v1, off, 32);
        }
        acce0[p] = v0;
        acce1[p] = v1;
    }

    if (lane == 0) {
#pragma unroll
        for (int k = 0; k < NZD; ++k) red[wave * 48 + k] = accz[k];
#pragma unroll
        for (int p = 0; p < NPD; ++p) {
            red[wave * 48 + 32 + p] = acce0[p];
            red[wave * 48 + 36 + p] = acce1[p];
        }
    }
    __syncthreads();

    // cross-wave reduce + direct ("eye" relation) gathers; write z_cat row
    if (tid < 40) {
        float s = 0.0f;
#pragma unroll
        for (int w = 0; w < 8; ++w) s += red[w * 48 + tid];

        float* out = zcat + (size_t)b * KDIM;
        if (tid < 32) {
            out[tid]      = fbase[(size_t)x * NZD + tid];   // z_context r=0 (eye)
            out[32 + tid] = s;                              // z_context r=1
        } else if (tid < 36) {
            const int p = tid - 32;
            out[64 + p] = e0base[(size_t)x * NPD + p];      // exo_c0 r=0
            out[68 + p] = s;                                // exo_c0 r=1
        } else {
            const int p = tid - 36;
            out[72 + p] = e1base[(size_t)x * NPD + p];      // exo_c1 r=0
            out[76 + p] = s;                                // exo_c1 r=1
        }
    }
}

// ---------------------------------------------------------------------------
// Phase 2: head GEMM  out = tanh(z_cat[4096,80] @ W_dyn^T[80,32] + b)
// via V_WMMA_F32_16X16X4_F32 (full fp32). 256 blocks (16-row tiles) x 2 waves
// (one per 16-wide N tile). K=80 -> 20 WMMA steps of K=4.
//
// A-frag (16x4 f32): lanes 0-15 hold M=lane, K={0,1}; lanes 16-31 hold
// M=lane-16, K={2,3}  -> aligned float2 load from the z_cat row.
// B-frag (4x16 f32): lanes 0-15 hold N=lane row K={0,1}; lanes 16-31 K={2,3}
//   B[k][n] = W_dyn[n][k] -> aligned float2 load from W_dyn row n.
// C/D (16x16 f32, 8 VGPRs): vgpr v, lanes 0-15 -> (M=v, N=lane),
// lanes 16-31 -> (M=v+8, N=lane-16).
// ---------------------------------------------------------------------------
__global__ __launch_bounds__(64)
void head_kernel(const float* __restrict__ zcat,  // [B, 80]
                 const float* __restrict__ W,     // [32, 80]
                 const float* __restrict__ bias,  // [32]
                 float*       __restrict__ out)   // [B, 32]
{
    const int tile  = blockIdx.x;            // 16-row tile of the batch
    const int lane  = threadIdx.x & 31;
    const int ntile = threadIdx.x >> 5;      // 0 or 1 (N tile)
    const int half  = lane >> 4;             // lane group (0: 0-15, 1: 16-31)
    const int l16   = lane & 15;

    const int arow  = tile * 16 + l16;       // A (z_cat) row for this lane
    const int kbase = half * 2;              // K sub-offset inside K4 group
    const int ncol  = ntile * 16 + l16;      // output column / W row

    const float* ap = zcat + (size_t)arow * KDIM;
    const float* bp = W    + (size_t)ncol * KDIM;

    v8f c = {0.f, 0.f, 0.f, 0.f, 0.f, 0.f, 0.f, 0.f};
#pragma unroll
    for (int k0 = 0; k0 < KDIM; k0 += 4) {
        v2f a = *(const v2f*)(ap + k0 + kbase);
        v2f bfrag = *(const v2f*)(bp + k0 + kbase);
        c = __builtin_amdgcn_wmma_f32_16x16x4_f32(
                /*neg_a=*/false, a, /*neg_b=*/false, bfrag,
                /*c_mod=*/(short)0, c, /*reuse_a=*/false, /*reuse_b=*/false);
    }

    const float bv = bias[ncol];
#pragma unroll
    for (int v = 0; v < 8; ++v) {
        const int row = tile * 16 + v + half * 8;
        out[(size_t)row * NZD + ncol] = tanhf(c[v] + bv);
    }
}

// ---------------------------------------------------------------------------
// Host-side launcher
// ---------------------------------------------------------------------------
extern "C" void kernel_launch(void* const* d_in, const int* in_sizes, int n_in,
                              void* d_out, int out_size, void* d_ws, size_t ws_size,
                              hipStream_t stream) {
    const float* relations = (const float*)d_in[0];  // [NX, 1, NX]
    const float* exogenous = (const float*)d_in[1];  // [NT, NX, NP]
    const float* factors   = (const float*)d_in[2];  // [NT, NX, NZ]
    const float* W_dyn     = (const float*)d_in[3];  // [NZ, 80]
    const float* b_dyn     = (const float*)d_in[4];  // [NZ]
    const int*   t_idx     = (const int*)d_in[5];    // [B]
    const int*   x_idx     = (const int*)d_in[6];    // [B]
    float*       out       = (float*)d_out;          // [B, NZ]

    float* zcat = (float*)d_ws;                      // [B, 80] = 1.31 MB

    zcat_kernel<<<BATCH, 256, 0, stream>>>(relations, exogenous, factors,
                                           t_idx, x_idx, zcat);
    head_kernel<<<BATCH / 16, 64, 0, stream>>>(zcat, W_dyn, b_dyn, out);
}